// SimpleUNetDepad_17145509446069
// MI455X (gfx1250) — compile-verified
//
#include <hip/hip_runtime.h>

typedef __attribute__((ext_vector_type(2))) float v2f;
typedef __attribute__((ext_vector_type(8))) float v8f;

#define K_OFF 27
#define M_RULES 200000
#define CIN 32
#define COUT 16
#define TILE 16
#define TILES_PER_K (M_RULES / TILE)        // 12500
#define BLOCKS_PER_K 98
#define WAVES_PER_BLOCK 8
#define WAVES_PER_K (BLOCKS_PER_K * WAVES_PER_BLOCK)  // 784

__global__ void zero_out_kernel(float* __restrict__ out, int n) {
  int i = blockIdx.x * blockDim.x + threadIdx.x;
  if (i < n) out[i] = 0.0f;
}

__global__ __launch_bounds__(256) void spconvT_wmma_kernel(
    const float* __restrict__ x,        // [N_SRC, CIN]
    const float* __restrict__ w,        // [K, CIN, COUT]
    const int*   __restrict__ src_idx,  // [K, M]
    const int*   __restrict__ dst_idx,  // [K, M]
    float*       __restrict__ out)      // [N_TGT, COUT]
{
  const int lane = threadIdx.x & 31;
  const int wave = threadIdx.x >> 5;
  const int k       = blockIdx.x / BLOCKS_PER_K;
  const int blkInK  = blockIdx.x % BLOCKS_PER_K;
  const int waveInK = blkInK * WAVES_PER_BLOCK + wave;

  const int hi = lane >> 4;   // 0: lanes 0-15, 1: lanes 16-31
  const int n  = lane & 15;   // column (N) / tile-row ownership

  // ---- Preload B = weight[k] (CIN x COUT) into WMMA 4x16 layout, 8 chunks.
  // Chunk kk covers K-rows 4kk..4kk+3.
  //   VGPR0 = { row 4kk+0 | row 4kk+2 },  VGPR1 = { row 4kk+1 | row 4kk+3 }
  const float* wk = w + (size_t)k * (CIN * COUT);
  v2f B[8];
#pragma unroll
  for (int kk = 0; kk < 8; ++kk) {
    const int r0 = 4 * kk + 2 * hi;
    B[kk].x = wk[(size_t)r0 * COUT + n];
    B[kk].y = wk[(size_t)(r0 + 1) * COUT + n];
  }

  const int* sk = src_idx + (size_t)k * M_RULES;
  const int* dk = dst_idx + (size_t)k * M_RULES;

  for (int t = waveInK; t < TILES_PER_K; t += WAVES_PER_K) {
    const int m0 = t * TILE;

    // Each lane owns rule-pair row m0+n (rows duplicated on lanes 16-31).
    const int si = sk[m0 + n];
    // Pre-scaled 32-bit element offset of this row's dst slot (scaled BEFORE
    // the shuffle: one multiply per row instead of per scattered element).
    const unsigned diOff = (unsigned)dk[m0 + n] * (unsigned)COUT;

    const float* xr = x + (size_t)si * CIN;

    v8f acc = {};
#pragma unroll
    for (int kk = 0; kk < 8; ++kk) {
      // A tile 16x4 f32: lane holds x[row, 4kk+2hi .. 4kk+2hi+1] -> exact ISA layout.
      const float2 f = *(const float2*)(xr + 4 * kk + 2 * hi);
      v2f a;
      a.x = f.x;
      a.y = f.y;
      // D = A x B + C  (v_wmma_f32_16x16x4_f32)
      acc = __builtin_amdgcn_wmma_f32_16x16x4_f32(
          /*neg_a=*/false, a, /*neg_b=*/false, B[kk],
          /*c_mod=*/(short)0, acc, /*reuse_a=*/false, /*reuse_b=*/false);
    }

    // Scatter-add: C layout -> lane holds rows r = v + 8*hi, col n, for v=0..7.
    // 32-bit unsigned offsets off a uniform base -> SGPR-base + voffset atomics.
#pragma unroll
    for (int v = 0; v < 8; ++v) {
      const int r = v + 8 * hi;
      const unsigned off = (unsigned)__shfl((int)diOff, r, 32) + (unsigned)n;
      atomicAdd(out + off, acc[v]);
    }
  }
}

extern "C" void kernel_launch(void* const* d_in, const int* in_sizes, int n_in,
                              void* d_out, int out_size, void* d_ws, size_t ws_size,
                              hipStream_t stream) {
  const float* x   = (const float*)d_in[0];
  const float* w   = (const float*)d_in[1];
  const int*   src = (const int*)d_in[2];
  const int*   dst = (const int*)d_in[3];
  // d_in[4] = n_tgt scalar; out_size already encodes n_tgt * COUT.
  float* out = (float*)d_out;

  zero_out_kernel<<<(out_size + 255) / 256, 256, 0, stream>>>(out, out_size);

  spconvT_wmma_kernel<<<K_OFF * BLOCKS_PER_K, 256, 0, stream>>>(x, w, src, dst, out);
}